// VirtualAdaptiveWeights_68350109549066
// MI455X (gfx1250) — compile-verified
//
#include <hip/hip_runtime.h>
#include <hip/hip_bf16.h>

// ---------------------------------------------------------------------------
// out[e] = x[src[e]] @ W[:256] + x[dst[e]] @ W[256:] + b
// Factored: (1) per-node projection GEMM via V_WMMA_F32_16X16X4_F32,
//           (2) per-edge gather+add from the tiny L2-resident proj table.
// ---------------------------------------------------------------------------

#define D_FEAT 256
#define NPROJ  16   // 8 heads via W[:256] packed next to 8 heads via W[256:]

typedef float v2f __attribute__((ext_vector_type(2)));
typedef float v8f __attribute__((ext_vector_type(8)));

// ---------------------------------------------------------------------------
// Kernel 1: proj[n][h]   = sum_k x[n][k] * W[k][h]        (h = 0..7)
//           proj[n][h+8] = sum_k x[n][k] * W[256+k][h]    (h = 0..7)
// One wave computes a 16x16 tile (16 nodes x 16 proj columns), accumulating
// 64 V_WMMA_F32_16X16X4_F32 steps over K=256.
// Block = 256 threads = 8 waves = 8 node tiles.
// ---------------------------------------------------------------------------
__global__ __launch_bounds__(256)
void node_proj_kernel(const float* __restrict__ x,
                      const float* __restrict__ W,   // [512][8] row-major
                      float* __restrict__ proj,      // [num_nodes][16]
                      int num_nodes)
{
    // WT[n][k] = combined weight, transposed: n<8 -> W[k][n], n>=8 -> W[256+k][n-8]
    __shared__ float WT[NPROJ * D_FEAT];             // 16 KB of 320 KB/WGP

    const int t = threadIdx.x;                        // 0..255 == k index
#pragma unroll
    for (int n = 0; n < NPROJ; ++n) {
        float v = (n < 8) ? W[t * 8 + n] : W[(D_FEAT + t) * 8 + (n - 8)];
        WT[n * D_FEAT + t] = v;
    }
    __syncthreads();

    const int wave = t >> 5;
    const int lane = t & 31;
    const int tile = blockIdx.x * 8 + wave;
    if (tile * 16 >= num_nodes) return;               // wave-uniform exit: EXEC
                                                      // stays all-ones for WMMA

    const int rc = lane & 15;                         // A: row-in-tile, B: column n
    const int kh = lane >> 4;                         // which K-pair of the 4

    // A operand: lane holds x[row][k + 2*kh], x[row][k + 2*kh + 1]
    const float* xrow = x + (size_t)(tile * 16 + rc) * D_FEAT + 2 * kh;
    // B operand: lane holds WT[n][k + 2*kh], WT[n][k + 2*kh + 1] (contiguous in LDS)
    const float* wrow = WT + rc * D_FEAT + 2 * kh;

    v8f acc = {};
#pragma unroll 4
    for (int s = 0; s < D_FEAT / 4; ++s) {            // 64 WMMA steps, K += 4
        v2f amat = *(const v2f*)(xrow + 4 * s);       // global_load_b64
        v2f bmat = *(const v2f*)(wrow + 4 * s);       // ds_load_b64
        // 8 args: (neg_a, A, neg_b, B, c_mod, C, reuse_a, reuse_b)
        acc = __builtin_amdgcn_wmma_f32_16x16x4_f32(
            false, amat, false, bmat, (short)0, acc, false, false);
    }

    // D layout: VGPR r -> row r (lanes 0-15) / row r+8 (lanes 16-31), col = lane&15
    const int mbase = kh * 8;
    float* op = proj + (size_t)(tile * 16 + mbase) * NPROJ + rc;
#pragma unroll
    for (int m = 0; m < 8; ++m)
        op[(size_t)m * NPROJ] = acc[m];
}

// ---------------------------------------------------------------------------
// Kernel 2: out[e][0:8] = proj[src[e]][0:8] + proj[dst[e]][8:16] + b
// proj is 3.2 MB -> fully L2-resident; gathers are L2 hits. b128 loads/stores.
// ---------------------------------------------------------------------------
__global__ __launch_bounds__(256)
void edge_gather_kernel(const int* __restrict__ ei,     // [2][num_edges]
                        const float* __restrict__ proj, // [num_nodes][16]
                        const float* __restrict__ bias, // [8]
                        float* __restrict__ out,        // [num_edges][8]
                        int num_edges)
{
    const int e = blockIdx.x * 256 + threadIdx.x;
    if (e >= num_edges) return;

    const int o = ei[e];
    const int d = ei[num_edges + e];

    const float4* pa = (const float4*)(proj + (size_t)o * NPROJ);      // a-part
    const float4* pc = (const float4*)(proj + (size_t)d * NPROJ + 8);  // c-part
    const float4* pb = (const float4*)bias;

    float4 a0 = pa[0], a1 = pa[1];
    float4 c0 = pc[0], c1 = pc[1];
    float4 b0 = pb[0], b1 = pb[1];

    float4 r0, r1;
    r0.x = a0.x + c0.x + b0.x;  r0.y = a0.y + c0.y + b0.y;
    r0.z = a0.z + c0.z + b0.z;  r0.w = a0.w + c0.w + b0.w;
    r1.x = a1.x + c1.x + b1.x;  r1.y = a1.y + c1.y + b1.y;
    r1.z = a1.z + c1.z + b1.z;  r1.w = a1.w + c1.w + b1.w;

    float4* po = (float4*)(out + (size_t)e * 8);
    po[0] = r0;
    po[1] = r1;
}

// ---------------------------------------------------------------------------
extern "C" void kernel_launch(void* const* d_in, const int* in_sizes, int n_in,
                              void* d_out, int out_size, void* d_ws, size_t ws_size,
                              hipStream_t stream)
{
    const float* x    = (const float*)d_in[0];   // [num_nodes, 256] f32
    const int*   ei   = (const int*)d_in[1];     // [2, num_edges] (integer input)
    const float* W    = (const float*)d_in[2];   // [512, 8] f32
    const float* bias = (const float*)d_in[3];   // [8] f32
    float*       out  = (float*)d_out;           // [num_edges, 8] f32
    float*       proj = (float*)d_ws;            // [num_nodes, 16] f32 scratch

    const int num_nodes = in_sizes[0] / D_FEAT;  // 50000
    const int num_edges = in_sizes[1] / 2;       // 500000

    const int tiles  = (num_nodes + 15) / 16;    // 3125 (exact, 16 | 50000)
    const int blocks = (tiles + 7) / 8;          // 8 waves (tiles) per block

    node_proj_kernel<<<blocks, 256, 0, stream>>>(x, W, proj, num_nodes);
    edge_gather_kernel<<<(num_edges + 255) / 256, 256, 0, stream>>>(
        ei, proj, bias, out, num_edges);
}